// SparseExpertLayer_66314295050802
// MI455X (gfx1250) — compile-verified
//
#include <hip/hip_runtime.h>
#include <math.h>

// Problem constants (from reference): x:(2,2048,512) V:(512,128,16) U:(128,512,16)
#define NTOK  4096
#define DDIM  512
#define MEXP  128
#define BRANK 16
#define MB    2048      // MEXP*BRANK
#define KTOP  4

typedef float v2f __attribute__((ext_vector_type(2)));
typedef float v8f __attribute__((ext_vector_type(8)));
typedef unsigned int u32x4 __attribute__((ext_vector_type(4)));
typedef int i32x4 __attribute__((ext_vector_type(4)));
typedef int i32x8 __attribute__((ext_vector_type(8)));

#if __has_builtin(__builtin_amdgcn_wmma_f32_16x16x4_f32)
#define HAVE_WMMA_F32 1
#endif
#if __has_builtin(__builtin_amdgcn_tensor_load_to_lds)
#define HAVE_TDM 1
#endif
#if __has_include(<hip/amd_detail/amd_gfx1250_TDM.h>)
#define TDM_6ARG 1   // therock-10.0 headers => 6-arg builtin flavor
#endif

// One 16x16x4 fp32 WMMA step: D = A(16x4) * B(4x16) + C
__device__ __forceinline__ v8f wmma_step(v8f c, v2f a, v2f b) {
#ifdef HAVE_WMMA_F32
  // 8 args: (neg_a, A, neg_b, B, c_mod, C, reuse_a, reuse_b)
  return __builtin_amdgcn_wmma_f32_16x16x4_f32(false, a, false, b, (short)0, c,
                                               false, false);
#else
  c[0] += a.x * b.x + a.y * b.y;  // placeholder so TU compiles; histogram wmma=0 flags it
  return c;
#endif
}

// Stage `tile_rows` consecutive rows of `row_elems` f32 (stride == row_elems, i.e.
// a contiguous block) from global into LDS. Prefer the Tensor Data Mover; fall back
// to a cooperative copy. Must be called by all 256 threads of the block; includes
// the trailing barrier.
__device__ __forceinline__ void stage_rows_to_lds(float* sdst, const float* tile_start,
                                                  unsigned row_elems, unsigned rows_total,
                                                  unsigned tile_rows) {
#ifdef HAVE_TDM
  if ((threadIdx.x >> 5) == 0) {  // one wave issues the TDM op
    unsigned long long ga = (unsigned long long)(uintptr_t)tile_start;
    unsigned lds_off = (unsigned)(uintptr_t)sdst;  // LDS aperture: low 32 bits = LDS offset
    u32x4 g0;
    g0[0] = 1u;                                         // count=1, user descriptor
    g0[1] = lds_off;                                    // lds_addr (bytes)
    g0[2] = (unsigned)(ga & 0xFFFFFFFFu);               // global_addr[31:0]
    g0[3] = (unsigned)((ga >> 32) & 0x01FFFFFFu)        // global_addr[56:32]
            | 0x80000000u;                              // type=2 ("image") in bits[127:126]
    i32x8 g1;
    g1[0] = (int)(2u << 16);                            // data_size=2 -> 4 bytes
    g1[1] = (int)((row_elems & 0xFFFFu) << 16);         // tensor_dim0[15:0]
    g1[2] = (int)((row_elems >> 16) | ((rows_total & 0xFFFFu) << 16));  // dim0 hi | dim1 lo
    g1[3] = (int)((rows_total >> 16) | ((row_elems & 0xFFFFu) << 16));  // dim1 hi | tile_dim0
    g1[4] = (int)(tile_rows & 0xFFFFu);                 // tile_dim1 (tile_dim2=0)
    g1[5] = (int)row_elems;                             // tensor_dim0_stride[31:0]
    g1[6] = 0;                                          // stride hi | dim1_stride lo
    g1[7] = 0;
    i32x4 z4 = {0, 0, 0, 0};
#ifdef TDM_6ARG
    i32x8 z8 = {0, 0, 0, 0, 0, 0, 0, 0};
    __builtin_amdgcn_tensor_load_to_lds(g0, g1, z4, z4, z8, 0);
#else
    __builtin_amdgcn_tensor_load_to_lds(g0, g1, z4, z4, 0);
#endif
    __builtin_amdgcn_s_wait_tensorcnt(0);
  }
#else
  (void)rows_total;
  unsigned n4 = (row_elems * tile_rows) >> 2;
  const float4* src = (const float4*)tile_start;
  float4* dst = (float4*)sdst;
  for (unsigned i = threadIdx.x; i < n4; i += 256) dst[i] = src[i];
#endif
  __syncthreads();
}

// ---------------------------------------------------------------- accumulators
__global__ void k_zero(float* __restrict__ acc) {
  int i = blockIdx.x * blockDim.x + threadIdx.x;
  if (i < 512) acc[i] = 0.0f;
}

// --------------------------------------------------- Vn^T: (MB, D) unit columns
__global__ void __launch_bounds__(256) k_norm_v(const float* __restrict__ V,
                                                float* __restrict__ VnT) {
  int col = blockIdx.x;         // col = m*B + b ; V flat index = d*MB + col
  int tid = threadIdx.x;
  float s = 0.0f;
  for (int d = tid; d < DDIM; d += 256) {
    float v = V[(size_t)d * MB + col];
    s += v * v;
  }
  __shared__ float red[256];
  red[tid] = s;
  __syncthreads();
  for (int off = 128; off > 0; off >>= 1) {
    if (tid < off) red[tid] += red[tid + off];
    __syncthreads();
  }
  float rn = 1.0f / sqrtf(red[0]);
  for (int d = tid; d < DDIM; d += 256)
    VnT[(size_t)col * DDIM + d] = V[(size_t)d * MB + col] * rn;
}

// --------------------------- U_eff^T: (MB, D), col = m*B+b, scaled by tanh(us)
__global__ void __launch_bounds__(256) k_norm_u(const float* __restrict__ U,
                                                const float* __restrict__ us,
                                                float* __restrict__ WT) {
  int col = blockIdx.x;
  int m = col >> 4, b = col & 15;
  int tid = threadIdx.x;
  const float* Ub = U + (size_t)m * DDIM * BRANK + b;  // stride BRANK over d
  float s = 0.0f;
  for (int d = tid; d < DDIM; d += 256) {
    float v = Ub[(size_t)d * BRANK];
    s += v * v;
  }
  __shared__ float red[256];
  red[tid] = s;
  __syncthreads();
  for (int off = 128; off > 0; off >>= 1) {
    if (tid < off) red[tid] += red[tid + off];
    __syncthreads();
  }
  float f = (1.0f / sqrtf(red[0])) * tanhf(us[col]);
  for (int d = tid; d < DDIM; d += 256)
    WT[(size_t)col * DDIM + d] = Ub[(size_t)d * BRANK] * f;
}

// ------------------------------- R(4096x2048) = X(4096x512) @ Vn(512x2048), WMMA
// Block = 8 waves sharing one 16-row A tile staged in LDS (TDM); each wave owns a
// 16x16 output tile (8 consecutive column tiles per block).
__global__ void __launch_bounds__(256) k_gemm_reprs(const float* __restrict__ X,
                                                    const float* __restrict__ VnT,
                                                    float* __restrict__ R) {
  __shared__ float sA[16 * DDIM];  // 32 KB
  int rowTile = blockIdx.x >> 4;   // 0..255
  int colGrp = blockIdx.x & 15;    // 0..15
  int tm = rowTile << 4;
  int w = threadIdx.x >> 5;
  int tn = ((colGrp << 3) + w) << 4;
  int lane = threadIdx.x & 31;
  int r = lane & 15;
  int ko = (lane >> 4) << 1;       // K sub-offset per ISA A/B fragment layout

  stage_rows_to_lds(sA, X + (size_t)tm * DDIM, DDIM, NTOK, 16);

  const float* ar = &sA[(size_t)r * DDIM + ko];
  const float* bp = VnT + (size_t)(tn + r) * DDIM + ko;
  __builtin_prefetch((const void*)bp, 0, 1);  // global_prefetch_b8
  v8f c = {0.f, 0.f, 0.f, 0.f, 0.f, 0.f, 0.f, 0.f};
#pragma unroll 8
  for (int k = 0; k < DDIM; k += 4) {
    v2f a = *(const v2f*)(ar + k);   // ds_load_b64
    v2f b = *(const v2f*)(bp + k);   // global_load_b64
    c = wmma_step(c, a, b);
  }
  int rowb = tm + ((lane >> 4) << 3);
  int colg = tn + r;
#pragma unroll
  for (int i = 0; i < 8; ++i)
    R[(size_t)(rowb + i) * MB + colg] = c[i];
}

// ------------- coherence penalty: sum |T T^T - I| over (MB x MB), T is (MB x D)
__global__ void __launch_bounds__(256) k_gram_pen(const float* __restrict__ VnT,
                                                  const float* __restrict__ WT,
                                                  float* __restrict__ penV,
                                                  float* __restrict__ penU) {
  __shared__ float sA[16 * DDIM];
  const float* T = blockIdx.y ? WT : VnT;
  float* pen = blockIdx.y ? penU : penV;
  int rowTile = blockIdx.x >> 4;   // 0..127
  int colGrp = blockIdx.x & 15;    // 0..15
  int ti = rowTile << 4;
  int w = threadIdx.x >> 5;
  int tj = ((colGrp << 3) + w) << 4;
  int lane = threadIdx.x & 31;
  int r = lane & 15;
  int ko = (lane >> 4) << 1;

  stage_rows_to_lds(sA, T + (size_t)ti * DDIM, DDIM, MB, 16);

  const float* ar = &sA[(size_t)r * DDIM + ko];
  const float* bp = T + (size_t)(tj + r) * DDIM + ko;
  __builtin_prefetch((const void*)bp, 0, 1);
  v8f c = {0.f, 0.f, 0.f, 0.f, 0.f, 0.f, 0.f, 0.f};
#pragma unroll 8
  for (int k = 0; k < DDIM; k += 4) {
    v2f a = *(const v2f*)(ar + k);
    v2f b = *(const v2f*)(bp + k);
    c = wmma_step(c, a, b);
  }
  int gi0 = ti + ((lane >> 4) << 3);
  int gj = tj + r;
  float s = 0.0f;
#pragma unroll
  for (int i = 0; i < 8; ++i) {
    float eye = ((gi0 + i) == gj) ? 1.0f : 0.0f;
    s += fabsf(c[i] - eye);
  }
  __shared__ float bsum;
  if (threadIdx.x == 0) bsum = 0.0f;
  __syncthreads();
  atomicAdd(&bsum, s);
  __syncthreads();
  if (threadIdx.x == 0) atomicAdd(pen, bsum);
}

// ------------------------------ per-token ||.||^2 over b, top-4 experts, gather
__global__ void __launch_bounds__(128) k_topk(const float* __restrict__ R,
                                              int* __restrict__ tidx,
                                              float* __restrict__ tval,
                                              float* __restrict__ rt,
                                              float* __restrict__ acc) {
  int n = blockIdx.x;
  int m = threadIdx.x;  // 0..127 == expert id
  __shared__ float l2[MEXP];
  __shared__ int sidx[KTOP];
  __shared__ float sval[KTOP];
  const float* row = R + (size_t)n * MB + m * BRANK;
  float s = 0.0f;
#pragma unroll
  for (int b = 0; b < BRANK; ++b) s += row[b] * row[b];
  l2[m] = s;
  __syncthreads();
  if (m == 0) {  // first-max tie-break matches jax top_k
    for (int k = 0; k < KTOP; ++k) {
      float best = -1.0f;
      int bi = 0;
      for (int i = 0; i < MEXP; ++i)
        if (l2[i] > best) { best = l2[i]; bi = i; }
      sidx[k] = bi; sval[k] = best;
      l2[bi] = -2.0f;
    }
  }
  __syncthreads();
  if (m < KTOP) {
    tidx[n * KTOP + m] = sidx[m];
    tval[n * KTOP + m] = sval[m];
    atomicAdd(&acc[sidx[m]], sval[m]);        // energy sum
    atomicAdd(&acc[MEXP + sidx[m]], 1.0f);    // select count
  }
  if (m < KTOP * BRANK) {
    int k = m >> 4, b = m & 15;
    rt[(size_t)n * (KTOP * BRANK) + m] = R[(size_t)n * MB + sidx[k] * BRANK + b];
  }
}

// ------------------- out = x + sum_k sum_b U_eff[m_k, d, b] * reprs_topk[k, b]
__global__ void __launch_bounds__(256) k_writes(const float* __restrict__ X,
                                                const float* __restrict__ WT,
                                                const int* __restrict__ tidx,
                                                const float* __restrict__ rt,
                                                float* __restrict__ out) {
  int n = blockIdx.x;
  int t = threadIdx.x;
  __shared__ float srt[KTOP * BRANK];
  __shared__ int sidx[KTOP];
  if (t < KTOP * BRANK) srt[t] = rt[(size_t)n * (KTOP * BRANK) + t];
  if (t < KTOP) sidx[t] = tidx[n * KTOP + t];
  __syncthreads();
#pragma unroll
  for (int j = 0; j < 2; ++j) {
    int d = t + j * 256;
    float a = X[(size_t)n * DDIM + d];
#pragma unroll
    for (int k = 0; k < KTOP; ++k) {
      const float* wp = WT + (size_t)sidx[k] * BRANK * DDIM + d;  // col m*B+b rows
#pragma unroll
      for (int b = 0; b < BRANK; ++b) a += wp[(size_t)b * DDIM] * srt[k * BRANK + b];
    }
    out[(size_t)n * DDIM + d] = a;
  }
}

// --------------------------------------------------------------- scalar tails
__global__ void __launch_bounds__(128) k_final(const float* __restrict__ acc,
                                               float* __restrict__ tail) {
  int m = threadIdx.x;
  float e = acc[m] * (1.0f / (float)NTOK);
  float sr = acc[MEXP + m] * (1.0f / (float)NTOK);
  __shared__ float red[MEXP];
  red[m] = e;
  __syncthreads();
  for (int off = 64; off > 0; off >>= 1) {
    if (m < off) red[m] += red[m + off];
    __syncthreads();
  }
  float total = fmaxf(red[0], 1e-12f);
  tail[1 + m] = e;                 // energy_per_expert
  tail[1 + MEXP + m] = e / total;  // relative_energy
  tail[1 + 2 * MEXP + m] = sr;     // select_rate
  if (m == 0) {
    float sq = sqrtf((float)DDIM);
    float inv = 1.0f / ((float)MB * (float)MB);
    float vpen = acc[256] * inv * sq;
    float upen = acc[257] * inv * sq;
    tail[0] = total;                          // total_energy
    tail[1 + 3 * MEXP + 0] = vpen;            // v_pen
    tail[1 + 3 * MEXP + 1] = upen;            // u_pen
    tail[1 + 3 * MEXP + 2] = 0.01f * vpen - total;  // aux_loss
  }
}

extern "C" void kernel_launch(void* const* d_in, const int* in_sizes, int n_in,
                              void* d_out, int out_size, void* d_ws, size_t ws_size,
                              hipStream_t stream) {
  (void)in_sizes; (void)n_in; (void)out_size; (void)ws_size;
  const float* x  = (const float*)d_in[0];   // (2,2048,512)
  const float* V  = (const float*)d_in[1];   // (512,128,16)
  const float* U  = (const float*)d_in[2];   // (128,512,16)
  const float* us = (const float*)d_in[3];   // (128,16)
  float* out = (float*)d_out;
  float* ws  = (float*)d_ws;

  // workspace layout (float offsets)
  float* acc  = ws;                          // [0,512): energy(128), count(128), penV, penU
  float* VnT  = ws + 512;                    // 2048*512
  float* WT   = ws + 512 + 1048576;          // 2048*512
  float* R    = ws + 2097664;                // 4096*2048
  int*   tidx = (int*)(ws + 10486272);       // 4096*4
  float* tval = ws + 10502656;               // 4096*4
  float* rt   = ws + 10519040;               // 4096*64

  k_zero<<<2, 256, 0, stream>>>(acc);
  k_norm_v<<<MB, 256, 0, stream>>>(V, VnT);
  k_norm_u<<<MB, 256, 0, stream>>>(U, us, WT);
  k_gemm_reprs<<<NTOK, 256, 0, stream>>>(x, VnT, R);
  k_gram_pen<<<dim3(2048, 2), 256, 0, stream>>>(VnT, WT, acc + 256, acc + 257);
  k_topk<<<NTOK, 128, 0, stream>>>(R, tidx, tval, rt, acc);
  k_writes<<<NTOK, 256, 0, stream>>>(x, WT, tidx, rt, out);
  k_final<<<1, 128, 0, stream>>>(acc, out + (size_t)NTOK * DDIM);
}